// StemExtraction_83580063580805
// MI455X (gfx1250) — compile-verified
//
#include <hip/hip_runtime.h>
#include <math.h>

#define BATCH 8
#define HH 1024
#define WW 1024
#define HW (HH * WW)
#define NTOT (BATCH * HW)
#define RADIUS 15.0f
#define THRESH_VOTES 0.01f
#define THRESH_PEAKS 0.2f
#define INV_AREA (1.0f / (3.14159265358979323846f * RADIUS * RADIUS))

typedef __attribute__((ext_vector_type(2))) float v2f;
typedef __attribute__((ext_vector_type(8))) float v8f;

// ---------------------------------------------------------------- zero votes
__global__ void zero_kernel(float* __restrict__ p, int n) {
    int i = blockIdx.x * blockDim.x + threadIdx.x;
    if (i < n) p[i] = 0.0f;
}

// ------------------------------------------------------- scatter-add voting
__global__ void scatter_kernel(const float* __restrict__ kp,
                               const float* __restrict__ off,
                               float* __restrict__ votes) {
    int i = blockIdx.x * blockDim.x + threadIdx.x;
    if (i >= NTOT) return;
    int b   = i / HW;
    int rem = i - b * HW;
    int y   = rem / WW;
    int x   = rem - y * WW;

    float w  = kp[i];
    float ox = off[(size_t)b * 2 * HW + rem];
    float oy = off[(size_t)b * 2 * HW + HW + rem];

    // jnp.round == round-half-to-even == __float2int_rn
    int vx = __float2int_rn((float)x + RADIUS * ox);
    int vy = __float2int_rn((float)y + RADIUS * oy);

    bool valid = (vx >= 0) && (vx < WW) && (vy >= 0) && (vy < HH) && (w >= THRESH_VOTES);
    if (valid) {
        // fold the pi*R^2 normalization into the scatter (box sum is linear)
        atomicAdd(&votes[(size_t)b * HW + (size_t)vy * WW + vx], w * INV_AREA);
    }
}

// ------------------------------------------- horizontal 15-wide box sum (LDS)
__global__ void hbox_kernel(const float* __restrict__ in, float* __restrict__ out) {
    __shared__ float s[256 + 14];
    int b  = blockIdx.z;
    int y  = blockIdx.y;
    int x0 = blockIdx.x * 256;
    const float* row = in + (size_t)b * HW + (size_t)y * WW;

    for (int i = threadIdx.x; i < 256 + 14; i += 256) {
        int gx = x0 - 7 + i;
        s[i] = (gx >= 0 && gx < WW) ? row[gx] : 0.0f;  // SAME => zero pad
    }
    __syncthreads();

    float acc = 0.0f;
#pragma unroll
    for (int k = 0; k < 15; ++k) acc += s[threadIdx.x + k];
    out[(size_t)b * HW + (size_t)y * WW + x0 + threadIdx.x] = acc;
}

// ------------------- vertical 15-tall box sum as banded-ones WMMA matmul.
// One wave32 computes one 16x16 output tile:
//   out(16x16) = sum_{j=0..7} Band_j(16x4) x In_j(4x16),  K window = rows y0-8..y0+23
// Band_j[m][k] = 1 iff (j*4+k) in [m+1, m+15]   (|input_row - out_row| <= 7)
// Out-of-image rows: address clamped (always-valid load), coefficient zeroed ->
// no exec-diverging loads. Interior/edge split is scalar (y0 from blockIdx only):
// a block's 8 waves cover 8 horizontally adjacent tiles (16 rows x 128 cols).
__global__ void __launch_bounds__(256) vbox_wmma_kernel(const float* __restrict__ in,
                                                        float* __restrict__ out) {
    int lane = threadIdx.x & 31;
    int wv   = threadIdx.x >> 5;

    int b    = blockIdx.x >> 9;          // 512 blocks per image
    int rest = blockIdx.x & 511;
    int ty   = rest >> 3;                // 64 tile-rows
    int txg  = rest & 7;                 // 8 tile-column groups of 8
    int y0   = ty * 16;                  // scalar (uniform across block)
    int x0   = (txg * 8 + wv) * 16;

    const float* base = in + (size_t)b * HW;
    int m     = lane & 15;
    int khalf = (lane >> 4) * 2;         // lanes 0-15 hold K0/K1, lanes 16-31 hold K2/K3
    int col   = x0 + m;

    v8f acc = {};
    if (y0 >= 8 && y0 + 24 <= HH) {
        // Fast path (62/64 tile-rows): all 32 window rows in-image.
        const float* p = base + (size_t)(y0 - 8) * WW + col;
#pragma unroll
        for (int j = 0; j < 8; ++j) {
            int k0 = j * 4 + khalf;
            v2f a;
            a.x = (k0     > m && k0     <= m + 15) ? 1.0f : 0.0f;
            a.y = (k0 + 1 > m && k0 + 1 <= m + 15) ? 1.0f : 0.0f;
            v2f bb;
            bb.x = p[(size_t)(k0)     * WW];
            bb.y = p[(size_t)(k0 + 1) * WW];
            acc = __builtin_amdgcn_wmma_f32_16x16x4_f32(false, a, false, bb,
                                                        (short)0, acc, false, false);
        }
    } else {
        // Edge path: clamp addresses, zero the band coefficient for invalid rows.
#pragma unroll
        for (int j = 0; j < 8; ++j) {
            int k0 = j * 4 + khalf;
            int r0 = y0 - 8 + k0;
            int r1 = r0 + 1;
            bool v0 = (r0 >= 0) && (r0 < HH);
            bool v1 = (r1 >= 0) && (r1 < HH);
            int r0c = min(max(r0, 0), HH - 1);
            int r1c = min(max(r1, 0), HH - 1);
            v2f a;
            a.x = (k0     > m && k0     <= m + 15 && v0) ? 1.0f : 0.0f;
            a.y = (k0 + 1 > m && k0 + 1 <= m + 15 && v1) ? 1.0f : 0.0f;
            v2f bb;
            bb.x = base[(size_t)r0c * WW + col];
            bb.y = base[(size_t)r1c * WW + col];
            acc = __builtin_amdgcn_wmma_f32_16x16x4_f32(false, a, false, bb,
                                                        (short)0, acc, false, false);
        }
    }

    // D layout: VGPR r -> M=r (lanes 0-15), M=r+8 (lanes 16-31)
    float* ob   = out + (size_t)b * HW;
    int   orow0 = y0 + (lane >> 4) * 8;
#pragma unroll
    for (int r = 0; r < 8; ++r) {
        ob[(size_t)(orow0 + r) * WW + col] = acc[r];
    }
}

// ------------------------------------------- horizontal 17-wide max (LDS)
__global__ void hmax_kernel(const float* __restrict__ in, float* __restrict__ out) {
    __shared__ float s[256 + 16];
    int b  = blockIdx.z;
    int y  = blockIdx.y;
    int x0 = blockIdx.x * 256;
    const float* row = in + (size_t)b * HW + (size_t)y * WW;

    for (int i = threadIdx.x; i < 256 + 16; i += 256) {
        int gx = x0 - 8 + i;
        s[i] = (gx >= 0 && gx < WW) ? row[gx] : -INFINITY;  // SAME => -inf pad
    }
    __syncthreads();

    float m = -INFINITY;
#pragma unroll
    for (int k = 0; k < 17; ++k) m = fmaxf(m, s[threadIdx.x + k]);
    out[(size_t)b * HW + (size_t)y * WW + x0 + threadIdx.x] = m;
}

// -------------------------- vertical 17-tall max + peak detection (isclose)
__global__ void vmax_peaks_kernel(const float* __restrict__ hmaxed,
                                  const float* __restrict__ votes,
                                  float* __restrict__ peaks) {
    int b = blockIdx.z;
    int y = blockIdx.y;
    int x = blockIdx.x * 256 + threadIdx.x;

    const float* colp = hmaxed + (size_t)b * HW + x;
    float m = -INFINITY;
#pragma unroll
    for (int k = -8; k <= 8; ++k) {
        int yy = y + k;
        if (yy >= 0 && yy < HH) m = fmaxf(m, colp[(size_t)yy * WW]);
    }

    size_t idx = (size_t)b * HW + (size_t)y * WW + x;
    float  v   = votes[idx];
    // jnp.isclose: |a-b| <= atol + rtol*|b|  (atol=1e-8, rtol=1e-5)
    bool close = fabsf(v - m) <= (1e-8f + 1e-5f * fabsf(m));
    peaks[idx] = (close && v > THRESH_PEAKS) ? 1.0f : 0.0f;
}

extern "C" void kernel_launch(void* const* d_in, const int* in_sizes, int n_in,
                              void* d_out, int out_size, void* d_ws, size_t ws_size,
                              hipStream_t stream) {
    const float* kp  = (const float*)d_in[0];  // (8,1,1024,1024) f32
    const float* off = (const float*)d_in[1];  // (8,2,1024,1024) f32

    float* out_votes = (float*)d_out;          // first 8M floats
    float* out_peaks = out_votes + NTOT;       // second 8M (bool peaks as 0/1 f32)

    float* votes_raw = (float*)d_ws;           // 32 MB scratch plane
    float* tmp       = votes_raw + NTOT;       // 32 MB scratch plane

    const int threads = 256;
    const int blocksN = (NTOT + threads - 1) / threads;
    dim3 grid_rows(WW / 256, HH, BATCH);

    zero_kernel<<<blocksN, threads, 0, stream>>>(votes_raw, NTOT);
    scatter_kernel<<<blocksN, threads, 0, stream>>>(kp, off, votes_raw);
    hbox_kernel<<<grid_rows, threads, 0, stream>>>(votes_raw, tmp);

    // 8 images * 512 blocks; each block = 8 waves = 8 adjacent 16x16 tiles
    vbox_wmma_kernel<<<BATCH * 512, 256, 0, stream>>>(tmp, out_votes);

    hmax_kernel<<<grid_rows, threads, 0, stream>>>(out_votes, votes_raw);
    vmax_peaks_kernel<<<grid_rows, threads, 0, stream>>>(votes_raw, out_votes, out_peaks);
}